// PConv_4097398800822
// MI455X (gfx1250) — compile-verified
//
#include <hip/hip_runtime.h>

typedef __attribute__((ext_vector_type(2))) float v2f;
typedef __attribute__((ext_vector_type(8))) float v8f;

#define NPTS   100000
#define KNBR   16
#define CIN    32
#define CADD   3
#define CMID   16
#define CTOT   35
#define OUTPP  (CTOT * CMID)   // 560 floats per point
#define RSTRIDE 48             // padded LDS row: covers tile-2 reads (cols 32..47)
#define WPB    8               // waves per block (256 threads, wave32)

__global__ __launch_bounds__(256) void pconv_wmma_kernel(
    const float* __restrict__ inF,   // [N, 32]
    const int*   __restrict__ nbr,   // [N, 16]
    const float* __restrict__ wnet,  // [N, 16, 16]  (k-major, m-minor)
    const float* __restrict__ addF,  // [N, 16, 3]
    float*       __restrict__ out)   // [N, 560]  (c-major, m-minor)
{
    __shared__ float smem[WPB * KNBR * RSTRIDE];   // 24 KB

    const int lane = threadIdx.x & 31;
    const int wave = threadIdx.x >> 5;
    int p = blockIdx.x * WPB + wave;               // grid is exact: 12500*8 = 100000
    p = __builtin_amdgcn_readfirstlane(p);         // force scalar base addressing

    float* feat = smem + wave * (KNBR * RSTRIDE);

    // ---- gather neighbor features into LDS: feat[k][c], c < 32 (coalesced 128B per k)
    const int* idxp = nbr + (size_t)p * KNBR;
    #pragma unroll
    for (int k = 0; k < KNBR; ++k) {
        int src = idxp[k];                         // wave-uniform -> s_load
        feat[k * RSTRIDE + lane] = inF[(size_t)src * CIN + lane];
    }
    // ---- additional features: feat[k][32+c], 48 elements total
    {
        const float* ap = addF + (size_t)p * (KNBR * CADD);
        int e = lane;
        if (e < KNBR * CADD)
            feat[(e / CADD) * RSTRIDE + CIN + (e % CADD)] = ap[e];
        e = lane + 32;
        if (e < KNBR * CADD)
            feat[(e / CADD) * RSTRIDE + CIN + (e % CADD)] = ap[e];
    }

    // ---- B fragments (weightnet) in WMMA 16x16x4 B-layout:
    //      lane (m = lane&15, h = lane>>4); chunk q: VGPR j = W[4q + 2h + j][m]
    const int m = lane & 15;
    const int h = lane >> 4;
    const float* Wp = wnet + (size_t)p * (KNBR * CMID);
    v2f bfrag[4];
    #pragma unroll
    for (int q = 0; q < 4; ++q) {
        bfrag[q].x = Wp[(4 * q + 2 * h + 0) * CMID + m];
        bfrag[q].y = Wp[(4 * q + 2 * h + 1) * CMID + m];
    }

    __syncthreads();   // make cross-lane LDS writes visible before A-fragment reads

    // ---- D(16x16) += A(16x4) x B(4x16), 4 k-chunks, 3 channel tiles
    v8f acc0 = {0.f, 0.f, 0.f, 0.f, 0.f, 0.f, 0.f, 0.f};
    v8f acc1 = acc0;
    v8f acc2 = acc0;
    #pragma unroll
    for (int q = 0; q < 4; ++q) {
        const int k0 = 4 * q + 2 * h;
        v2f a;
        a.x = feat[(k0 + 0) * RSTRIDE + 0 * 16 + m];
        a.y = feat[(k0 + 1) * RSTRIDE + 0 * 16 + m];
        acc0 = __builtin_amdgcn_wmma_f32_16x16x4_f32(false, a, false, bfrag[q],
                                                     (short)0, acc0, false, false);
        a.x = feat[(k0 + 0) * RSTRIDE + 1 * 16 + m];
        a.y = feat[(k0 + 1) * RSTRIDE + 1 * 16 + m];
        acc1 = __builtin_amdgcn_wmma_f32_16x16x4_f32(false, a, false, bfrag[q],
                                                     (short)0, acc1, false, false);
        a.x = feat[(k0 + 0) * RSTRIDE + 2 * 16 + m];
        a.y = feat[(k0 + 1) * RSTRIDE + 2 * 16 + m];
        acc2 = __builtin_amdgcn_wmma_f32_16x16x4_f32(false, a, false, bfrag[q],
                                                     (short)0, acc2, false, false);
    }

    // ---- store D: lane l, VGPR r -> out row c = tile*16 + r + 8h, col m
    float* op = out + (size_t)p * OUTPP;
    #pragma unroll
    for (int r = 0; r < 8; ++r) op[(0  + r + 8 * h) * CMID + m] = acc0[r];
    #pragma unroll
    for (int r = 0; r < 8; ++r) op[(16 + r + 8 * h) * CMID + m] = acc1[r];
    if (h == 0) {                                   // tile 2: only rows 32..34 are real
        #pragma unroll
        for (int r = 0; r < 3; ++r) op[(32 + r) * CMID + m] = acc2[r];
    }
}

extern "C" void kernel_launch(void* const* d_in, const int* in_sizes, int n_in,
                              void* d_out, int out_size, void* d_ws, size_t ws_size,
                              hipStream_t stream) {
    const float* inF  = (const float*)d_in[0];   // input_features  [1,100000,32] f32
    const int*   nbr  = (const int*)  d_in[1];   // neighbor_inds   [1,100000,16] int
    const float* wnet = (const float*)d_in[2];   // weightnet       [1,100000,16,16] f32
    const float* addF = (const float*)d_in[3];   // additional      [1,100000,16,3] f32
    float* out = (float*)d_out;                  // [1,100000,560] f32

    dim3 grid(NPTS / WPB);   // 12500 blocks * 8 waves = 100000 points exactly
    dim3 block(256);
    hipLaunchKernelGGL(pconv_wmma_kernel, grid, block, 0, stream,
                       inF, nbr, wnet, addF, out);
}